// DilatedMultiheadSelfAttentionWithRelativePositionalEmbedding_15590731284951
// MI455X (gfx1250) — compile-verified
//
#include <hip/hip_runtime.h>
#include <hip/hip_bf16.h>

// ---------------------------------------------------------------------------
// Dilated multi-head self-attention w/ relative positional embedding (MI455X)
//   Stage 1: Q/K/V projection GEMMs, V_WMMA_F32_16X16X4_F32, 2x2 register
//            blocking, double-buffered GLOBAL_LOAD_ASYNC_TO_LDS staging.
//   Stage 2: 5-wide dilated-window attention epilogue (wave-per-head).
// ---------------------------------------------------------------------------

#define BATCH    4
#define SEQ      4096
#define DMODEL   512
#define NHEADS   8
#define HEADDIM  64
#define WLEN     5
#define MTOT     (BATCH * SEQ)        // 16384 rows in the projection GEMMs

#define MBLK     128                  // block output tile M
#define NBLK     64                   // block output tile N
#define KBLK     16                   // K step per stage
#define LXS      17                   // padded LDS row stride (floats)
#define NTILE    (DMODEL / KBLK)      // 32 K-steps

typedef float v2f __attribute__((ext_vector_type(2)));
typedef float v8f __attribute__((ext_vector_type(8)));

// --- CDNA5 async global->LDS copies (ASYNCcnt-tracked, bypass VGPRs) -------
// VDST operand = LDS byte offset; generic LDS pointers truncate to the LDS
// offset in their low 32 bits (ISA 10.2 aperture rules), so a u32 cast works.
__device__ __forceinline__ void async_ld_b128(const float* g, const float* l) {
    asm volatile("global_load_async_to_lds_b128 %0, %1, off"
                 :: "v"((unsigned)(uintptr_t)l),
                    "v"((unsigned long long)(uintptr_t)g)
                 : "memory");
}
__device__ __forceinline__ void async_ld_b32(const float* g, const float* l) {
    asm volatile("global_load_async_to_lds_b32 %0, %1, off"
                 :: "v"((unsigned)(uintptr_t)l),
                    "v"((unsigned long long)(uintptr_t)g)
                 : "memory");
}

// ===========================================================================
// Stage 1: C = X @ W + b for W in {Wq, Wk, Wv}  (blockIdx.z selects matrix)
//   M=16384, K=512, N=512.  Block: 256 threads = 8 wave32s, 128x64 tile.
//   Wave (wr,wc) owns a 32x32 region = 4 accumulators (2x2 of 16x16), so a
//   fragment set {a0,a1,b0,b1} feeds 4 WMMAs (2 LDS dwords per WMMA).
// ===========================================================================
__global__ __launch_bounds__(256) void qkv_gemm(
    const float* __restrict__ X,
    const float* __restrict__ Wq, const float* __restrict__ bq,
    const float* __restrict__ Wk, const float* __restrict__ bk,
    const float* __restrict__ Wv, const float* __restrict__ bv,
    float* __restrict__ Q, float* __restrict__ K, float* __restrict__ V)
{
    const float* Wm;
    const float* bm;
    float*       C;
    if (blockIdx.z == 0)      { Wm = Wq; bm = bq; C = Q; }
    else if (blockIdx.z == 1) { Wm = Wk; bm = bk; C = K; }
    else                      { Wm = Wv; bm = bv; C = V; }

    const int n0   = blockIdx.x * NBLK;
    const int m0   = blockIdx.y * MBLK;
    const int tid  = threadIdx.x;
    const int lane = tid & 31;
    const int wv   = tid >> 5;        // 0..7
    const int wr   = wv & 3;          // M sub-block (32 rows each)
    const int wc   = wv >> 2;         // N sub-block (32 cols each)
    const int hi   = lane >> 4;       // lane half (0/1)
    const int lo   = lane & 15;       // lane within half

    // lx : A tile [m][k], stride 17  (async B128 chunks stay inside a row)
    // lwt: B tile transposed [n][k], stride 17 -> B fragment = adjacent pair,
    //      odd stride -> conflict-free across the 64 LDS banks.
    __shared__ __align__(16) float lx [2][MBLK * LXS];
    __shared__ __align__(16) float lwt[2][NBLK * LXS];

    v8f c00 = {}, c01 = {}, c10 = {}, c11 = {};

    // 6 async instructions per wave per tile: 2x B128 (X) + 4x B32 (W^T scatter)
    auto issue = [&](int s, int kk) {
#pragma unroll
        for (int i = 0; i < 2; ++i) {
            const int e = tid + 256 * i;          // 512 chunks of 4 floats
            const int r = e >> 2, c4 = (e & 3) * 4;
            async_ld_b128(X + (size_t)(m0 + r) * DMODEL + kk + c4,
                          &lx[s][r * LXS + c4]);
        }
#pragma unroll
        for (int i = 0; i < 4; ++i) {
            const int e = tid + 256 * i;          // 1024 scalars, transposed
            const int r = e >> 6, c = e & 63;
            async_ld_b32(Wm + (size_t)(kk + r) * DMODEL + n0 + c,
                         &lwt[s][c * LXS + r]);
        }
    };

    issue(0, 0);
    for (int it = 0; it < NTILE; ++it) {
        const int cur = it & 1;
        if (it + 1 < NTILE) {
            issue(cur ^ 1, (it + 1) * KBLK);      // prefetch next K tile
            asm volatile("s_wait_asynccnt 0x6" ::: "memory"); // cur tile done
        } else {
            asm volatile("s_wait_asynccnt 0x0" ::: "memory");
        }
        __syncthreads();

        const float* xb = lx[cur];
        const float* wb = lwt[cur];
#pragma unroll
        for (int sk = 0; sk < KBLK; sk += 4) {
            // A (16x4): lane l -> X[m = l%16, K = sk + v + 2*(l/16)]
            // B (4x16): lane l -> W[K = sk + v + 2*(l/16), n = l%16]
            const int kr = sk + 2 * hi;
            const float* pa0 = xb + (32 * wr + lo) * LXS + kr;
            const float* pa1 = pa0 + 16 * LXS;
            const float* pb0 = wb + (32 * wc + lo) * LXS + kr;
            const float* pb1 = pb0 + 16 * LXS;
            v2f a0; a0.x = pa0[0]; a0.y = pa0[1];
            v2f a1; a1.x = pa1[0]; a1.y = pa1[1];
            v2f b0; b0.x = pb0[0]; b0.y = pb0[1];
            v2f b1; b1.x = pb1[0]; b1.y = pb1[1];
            c00 = __builtin_amdgcn_wmma_f32_16x16x4_f32(
                      false, a0, false, b0, (short)0, c00, false, false);
            c01 = __builtin_amdgcn_wmma_f32_16x16x4_f32(
                      false, a0, false, b1, (short)0, c01, false, false);
            c10 = __builtin_amdgcn_wmma_f32_16x16x4_f32(
                      false, a1, false, b0, (short)0, c10, false, false);
            c11 = __builtin_amdgcn_wmma_f32_16x16x4_f32(
                      false, a1, false, b1, (short)0, c11, false, false);
        }
        __syncthreads();
    }

    // Epilogue: C/D layout -> VGPR r, lane l: (M = r + 8*(l/16), N = l%16).
#pragma unroll
    for (int r = 0; r < 8; ++r) {
        const int mA = m0 + 32 * wr + r + 8 * hi;
        const int mB = mA + 16;
        const int nA = n0 + 32 * wc + lo;
        const int nB = nA + 16;
        C[(size_t)mA * DMODEL + nA] = c00[r] + bm[nA];
        C[(size_t)mA * DMODEL + nB] = c01[r] + bm[nB];
        C[(size_t)mB * DMODEL + nA] = c10[r] + bm[nA];
        C[(size_t)mB * DMODEL + nB] = c11[r] + bm[nB];
    }
}

// ===========================================================================
// Stage 2: dilated-window attention.
//   One block per (b,t); wave = head h.  Lane owns dims {2*lane, 2*lane+1}.
//   kv_roll window w == row (t + (shift_h + w)*dil - p), zero & masked if OOB
//   (jnp.roll's circular wrap always lands in the zero pad -> clamp is exact).
// ===========================================================================
__global__ __launch_bounds__(256) void attn_win(
    const float* __restrict__ Q, const float* __restrict__ Km,
    const float* __restrict__ Vm, const float* __restrict__ Er,
    const int* __restrict__ layer_p,
    float* __restrict__ out, float* __restrict__ attn_out)
{
    const int b    = blockIdx.x / SEQ;
    const int t    = blockIdx.x % SEQ;
    const int tid  = threadIdx.x;
    const int h    = tid >> 5;
    const int lane = tid & 31;

    const int dil = 1 << layer_p[0];
    const int pad = dil * (WLEN / 2);
    int shift;
    if (h < 4)       shift = 0;
    else if (h == 4) shift = -2;
    else if (h == 5) shift = -1;
    else if (h == 6) shift = 1;
    else             shift = 2;

    const int    d0   = 2 * lane;
    const size_t base = ((size_t)(b * SEQ + t)) * DMODEL + h * HEADDIM + d0;
    const float2 q2   = *(const float2*)(Q + base);

    float qk[WLEN], qe[WLEN];
    int   idxs[WLEN];
#pragma unroll
    for (int w = 0; w < WLEN; ++w) {
        const int  idx = t + (shift + w) * dil - pad;
        const bool ok  = (idx >= 0) && (idx < SEQ);
        idxs[w] = ok ? idx : -1;
        float2 k2 = make_float2(0.f, 0.f);
        if (ok)
            k2 = *(const float2*)(Km + ((size_t)(b * SEQ + idx)) * DMODEL +
                                  h * HEADDIM + d0);
        float pk = q2.x * k2.x + q2.y * k2.y;
        float pe = q2.x * Er[(h * HEADDIM + d0) * WLEN + w] +
                   q2.y * Er[(h * HEADDIM + d0 + 1) * WLEN + w];
#pragma unroll
        for (int m = 16; m >= 1; m >>= 1) {
            pk += __shfl_xor(pk, m, 32);
            pe += __shfl_xor(pe, m, 32);
        }
        qk[w] = pk;
        qe[w] = pe;
    }

    // logits = (qk + qEr)/sqrt(hd) + (qk==0)*-1e4 ; softmax over the window
    float lg[WLEN], mx = -3.0e38f;
#pragma unroll
    for (int w = 0; w < WLEN; ++w) {
        float l = (qk[w] + qe[w]) * 0.125f;      // 1/sqrt(64)
        if (qk[w] == 0.0f) l -= 10000.0f;
        lg[w] = l;
        mx = fmaxf(mx, l);
    }
    float sum = 0.f, aw[WLEN];
#pragma unroll
    for (int w = 0; w < WLEN; ++w) { aw[w] = __expf(lg[w] - mx); sum += aw[w]; }
    const float inv = 1.0f / sum;
#pragma unroll
    for (int w = 0; w < WLEN; ++w) aw[w] *= inv;

    // out = sum_w attn[w] * V_row(w)
    float2 o = make_float2(0.f, 0.f);
#pragma unroll
    for (int w = 0; w < WLEN; ++w) {
        if (idxs[w] >= 0) {
            const float2 v2 = *(const float2*)(
                Vm + ((size_t)(b * SEQ + idxs[w])) * DMODEL + h * HEADDIM + d0);
            o.x += aw[w] * v2.x;
            o.y += aw[w] * v2.y;
        }
    }
    *(float2*)(out + base) = o;

    if (lane == 0) {
        float* ap = attn_out + (((size_t)b * NHEADS + h) * SEQ + t) * WLEN;
#pragma unroll
        for (int w = 0; w < WLEN; ++w) ap[w] = aw[w];
    }
}

// ===========================================================================
extern "C" void kernel_launch(void* const* d_in, const int* in_sizes, int n_in,
                              void* d_out, int out_size, void* d_ws,
                              size_t ws_size, hipStream_t stream)
{
    const float* X     = (const float*)d_in[0];
    const float* Wq    = (const float*)d_in[1];
    const float* bq    = (const float*)d_in[2];
    const float* Wk    = (const float*)d_in[3];
    const float* bk    = (const float*)d_in[4];
    const float* Wv    = (const float*)d_in[5];
    const float* bv    = (const float*)d_in[6];
    const float* Er    = (const float*)d_in[7];
    const int*   layer = (const int*)d_in[8];

    float* out = (float*)d_out;
    float* Q   = (float*)d_ws;                       // 3 x 32 MB scratch
    float* K   = Q + (size_t)MTOT * DMODEL;
    float* V   = K + (size_t)MTOT * DMODEL;
    float* attn_out = out + (size_t)MTOT * DMODEL;   // second tuple output

    dim3 g1(DMODEL / NBLK, MTOT / MBLK, 3);
    qkv_gemm<<<g1, 256, 0, stream>>>(X, Wq, bq, Wk, bk, Wv, bv, Q, K, V);

    attn_win<<<dim3(MTOT), 256, 0, stream>>>(Q, K, V, Er, layer, out, attn_out);
}